// LTCCell_11458972746384
// MI455X (gfx1250) — compile-verified
//
#include <hip/hip_runtime.h>

// ---------------------------------------------------------------------------
// LTC cell on gfx1250 (MI455X).
//   out[b,h] after 4 implicit-Euler unfolds of
//   resp = sens + sum_p sigmoid(sig*(state-mu))*W
// sigmoid(x) = 0.5*(1+tanh(0.5x)) -> single v_tanh_f32 per synapse.
// 272KB static LDS: fused (a',b',w') float4 params + ping-pong transposed state.
// Input/state transpose staging via global_load_async_to_lds_b32 (ASYNCcnt).
// ---------------------------------------------------------------------------

constexpr int Hn  = 128;   // hidden dim
constexpr int Sn  = 64;    // sensory dim
constexpr int BT  = 16;    // batch rows per block
constexpr int BPT = 4;     // batch rows per thread
constexpr int NT  = 512;   // threads per block = 128 h * 4 b-groups (16 waves)
constexpr int UNF = 4;     // ODE unfolds
#define LTC_EPS 1e-8f
#define LOG2E   1.4426950408889634f
#define USE_ASYNC_LDS 1

#if __has_builtin(__builtin_amdgcn_tanhf)
  #define HAVE_TANH 1
#elif __has_builtin(__builtin_amdgcn_tanh_f32)
  #define HAVE_TANH 2
#else
  #define HAVE_TANH 0
#endif

#if defined(__HIP_DEVICE_COMPILE__)
// Device-pass-only invariant: round-2 feedback proved the tanh builtin exists
// for gfx1250 (device job compiled clean; only the unguarded host pass failed).
static_assert(HAVE_TANH != 0, "device pass lost the v_tanh builtin");
#endif

__device__ __forceinline__ float fast_tanh(float x) {
#if HAVE_TANH == 1
  return __builtin_amdgcn_tanhf(x);
#elif HAVE_TANH == 2
  return __builtin_amdgcn_tanh_f32(x);
#else
  return 0.0f;
#endif
}

// One gated synapse: acc += w' * g(a'*x + b')
__device__ __forceinline__ float gate_acc(float a, float b, float w, float x, float acc) {
#if HAVE_TANH
  return __builtin_fmaf(w, fast_tanh(__builtin_fmaf(a, x, b)), acc);
#else
  // host-pass parse fallback (never executed on device: HAVE_TANH != 0 there)
  float e = __builtin_amdgcn_exp2f(__builtin_fmaf(a, x, b));
  float g = __builtin_amdgcn_rcpf(1.0f + e);
  return __builtin_fmaf(w, g, acc);
#endif
}

__device__ __forceinline__ float4 xform(float sig, float mu, float w) {
#if HAVE_TANH
  return make_float4(0.5f * sig, -0.5f * sig * mu, 0.5f * w, 0.0f);
#else
  return make_float4(-LOG2E * sig, LOG2E * sig * mu, w, 0.0f);
#endif
}

// --- CDNA5 async global->LDS copy (per-lane global addr AND per-lane LDS addr,
//     so the transpose scatter happens inside the DMA). Tracked by ASYNCcnt.
__device__ __forceinline__ void async_ld_b32(void* lds_ptr, const float* gptr) {
#if USE_ASYNC_LDS
  unsigned lds_off = (unsigned)(size_t)lds_ptr;  // low 32 bits of generic LDS addr
  asm volatile("global_load_async_to_lds_b32 %0, %1, off"
               :: "v"(lds_off), "v"(gptr) : "memory");
#else
  *(float*)lds_ptr = *gptr;
#endif
}
__device__ __forceinline__ void wait_async() {
#if USE_ASYNC_LDS
  asm volatile("s_wait_asynccnt 0x0" ::: "memory");
#endif
}

__global__ __launch_bounds__(NT)
void ltc_cell_kernel(const float* __restrict__ inp,     // [B,S]
                     const float* __restrict__ hid,     // [B,H]
                     const float* __restrict__ s_mu,    // [S,H]
                     const float* __restrict__ s_sig,   // [S,H]
                     const float* __restrict__ s_W,     // [S,H]
                     const float* __restrict__ h_mu,    // [H,H]
                     const float* __restrict__ h_sig,   // [H,H]
                     const float* __restrict__ h_W,     // [H,H]
                     const float* __restrict__ w_tau,   // [H]
                     const float* __restrict__ w_A,     // [H]
                     const float* __restrict__ elapsed, // [1]
                     float* __restrict__ out)           // [B,H]
{
  // 272 KB static LDS (CDNA5: up to 320 KB per workgroup)
  __shared__ __align__(16) float4 sm_abw[Hn * Hn];     // 256 KB: (a',b',w',0)
  __shared__ __align__(16) float  sm_st[2][Hn * BT];   //  16 KB: transposed state, ping-pong

  const int tid = threadIdx.x;
  const int h   = tid & (Hn - 1);   // 0..127 (wave32: 32 consecutive h per wave)
  const int g   = tid >> 7;         // 0..3 b-group
  const int bt0 = g * BPT;
  const int b0  = blockIdx.x * BT;

  // ---- Phase 1 staging: input tile transposed via async DMA + sensory params
  float* inT = &sm_st[0][0];        // [S][BT], 4 KB inside state buffer
  for (int i = tid; i < Sn * BT; i += NT) {
    int bt = i >> 6;                // i = bt*Sn + s, s fast -> coalesced global side
    int s  = i & (Sn - 1);
    async_ld_b32(&inT[s * BT + bt], inp + (b0 + bt) * Sn + s);
  }
  for (int i = tid; i < Sn * Hn; i += NT) {
    sm_abw[i] = xform(s_sig[i], s_mu[i], s_W[i]);
  }
  wait_async();
  __syncthreads();

  // ---- Phase 1 compute: sens[b,h] for this thread's 4 b rows
  float acc0, acc1, acc2, acc3;
#if HAVE_TANH
  {
    float base = 0.0f;              // 0.5*sum_s W[s,h]
    for (int s = 0; s < Sn; ++s) base += sm_abw[s * Hn + h].z;
    acc0 = acc1 = acc2 = acc3 = base;
  }
#else
  acc0 = acc1 = acc2 = acc3 = 0.0f;
#endif
#pragma unroll 4
  for (int s = 0; s < Sn; ++s) {
    const float4 abw = sm_abw[s * Hn + h];
    const float4 xv  = *(const float4*)&inT[s * BT + bt0]; // uniform broadcast in wave
    acc0 = gate_acc(abw.x, abw.y, abw.z, xv.x, acc0);
    acc1 = gate_acc(abw.x, abw.y, abw.z, xv.y, acc1);
    acc2 = gate_acc(abw.x, abw.y, abw.z, xv.z, acc2);
    acc3 = gate_acc(abw.x, abw.y, abw.z, xv.w, acc3);
  }
  __syncthreads();   // done reading sensory LDS regions

  // ---- Phase 2 staging: recurrent params + initial state transposed (async DMA)
  for (int i = tid; i < Hn * Hn; i += NT) {
    sm_abw[i] = xform(h_sig[i], h_mu[i], h_W[i]);
  }
  for (int i = tid; i < Hn * BT; i += NT) {
    int bt = i >> 7;                // i = bt*Hn + hp, hp fast -> coalesced global side
    int hp = i & (Hn - 1);
    async_ld_b32(&sm_st[0][hp * BT + bt], hid + (b0 + bt) * Hn + hp);
  }
  wait_async();
  __syncthreads();

#if HAVE_TANH
  {
    float base = 0.0f;              // 0.5*sum_p W[p,h]
    for (int p = 0; p < Hn; ++p) base += sm_abw[p * Hn + h].z;
    acc0 += base; acc1 += base; acc2 += base; acc3 += base;
  }
#endif

  // per-h constants of the ODE step
  const float dt  = elapsed[0] * (1.0f / (float)UNF);
  const float dtA = dt * w_A[h];
  const float c0  = 1.0f + dt * w_tau[h] + LTC_EPS;

  // ---- Unfold loop
  float4 ns = make_float4(0.f, 0.f, 0.f, 0.f);
  for (int u = 0; u < UNF; ++u) {
    const float* st  = &sm_st[u & 1][0];
    float*       stn = &sm_st[(u & 1) ^ 1][0];

    float r0 = acc0, r1 = acc1, r2 = acc2, r3 = acc3;
#pragma unroll 4
    for (int p = 0; p < Hn; ++p) {
      const float4 abw = sm_abw[p * Hn + h];                // ds_load_b128, dense
      const float4 sv  = *(const float4*)&st[p * BT + bt0]; // uniform broadcast
      r0 = gate_acc(abw.x, abw.y, abw.z, sv.x, r0);
      r1 = gate_acc(abw.x, abw.y, abw.z, sv.y, r1);
      r2 = gate_acc(abw.x, abw.y, abw.z, sv.z, r2);
      r3 = gate_acc(abw.x, abw.y, abw.z, sv.w, r3);
    }
    const float4 so = *(const float4*)&st[h * BT + bt0];
    ns.x = __builtin_fmaf(dtA, r0, so.x) / __builtin_fmaf(dt, r0, c0);
    ns.y = __builtin_fmaf(dtA, r1, so.y) / __builtin_fmaf(dt, r1, c0);
    ns.z = __builtin_fmaf(dtA, r2, so.z) / __builtin_fmaf(dt, r2, c0);
    ns.w = __builtin_fmaf(dtA, r3, so.w) / __builtin_fmaf(dt, r3, c0);
    *(float4*)&stn[h * BT + bt0] = ns;
    __syncthreads();
  }

  // ---- Output straight from registers, coalesced over h
  float* o = &out[(b0 + bt0) * Hn + h];
  o[0 * Hn] = ns.x;
  o[1 * Hn] = ns.y;
  o[2 * Hn] = ns.z;
  o[3 * Hn] = ns.w;
}

extern "C" void kernel_launch(void* const* d_in, const int* in_sizes, int n_in,
                              void* d_out, int out_size, void* d_ws, size_t ws_size,
                              hipStream_t stream) {
  const float* inp     = (const float*)d_in[0];
  const float* hid     = (const float*)d_in[1];
  const float* s_mu    = (const float*)d_in[2];
  const float* s_sig   = (const float*)d_in[3];
  const float* s_W     = (const float*)d_in[4];
  const float* h_mu    = (const float*)d_in[5];
  const float* h_sig   = (const float*)d_in[6];
  const float* h_W     = (const float*)d_in[7];
  const float* w_tau   = (const float*)d_in[8];
  const float* w_A     = (const float*)d_in[9];
  const float* elapsed = (const float*)d_in[10];
  float* out = (float*)d_out;

  const int B = in_sizes[0] / Sn;     // 8192
  dim3 grid(B / BT), block(NT);
  ltc_cell_kernel<<<grid, block, 0, stream>>>(inp, hid, s_mu, s_sig, s_W,
                                              h_mu, h_sig, h_W, w_tau, w_A,
                                              elapsed, out);
}